// GIN_60017872994746
// MI455X (gfx1250) — compile-verified
//
#include <hip/hip_runtime.h>
#include <stdint.h>

#define NODES  100000
#define EDGES  1600000
#define CH     128
#define KIT    5
#define NG     256
#define NCLS   10
#define FDIM   (CH * KIT)   // 640
#define BN_EPS 1e-5f

typedef __attribute__((ext_vector_type(2))) float v2f;
typedef __attribute__((ext_vector_type(8))) float v8f;

// ---------------------------------------------------------------------------
// CDNA5 async global->LDS staging (ASYNCcnt-tracked). 16B per issue.
// VDST operand = VGPR holding the wave-relative LDS byte address
// (= low 32 bits of the generic shared-memory pointer).
// ---------------------------------------------------------------------------
__device__ __forceinline__ void async_ld16(unsigned lds_byte_off, const float* gp) {
    asm volatile("global_load_async_to_lds_b128 %0, %1, off"
                 :: "v"(lds_byte_off), "v"(gp)
                 : "memory");
}
__device__ __forceinline__ void wait_async0() {
    asm volatile("s_wait_asynccnt 0x0" ::: "memory");
}

// ---------------------------------------------------------------------------
// One-time weight repack into WMMA-B fragment order.
// Fragment element idx = ((ct*(K/4) + k4)*32 + lane), each a v2f:
//   n = ct*16 + (lane&15),  k = k4*4 + 2*(lane>>4),  {W[k][n], W[k+1][n]}
// Consumed as one coalesced global_load_b64 per WMMA step.
// ---------------------------------------------------------------------------
__global__ void repack_B(const float* __restrict__ W, float* __restrict__ P,
                         int K, int N) {
    int idx = blockIdx.x * 256 + threadIdx.x;
    int total = (N >> 4) * (K >> 2) * 32;
    if (idx >= total) return;
    int lane = idx & 31;
    int t    = idx >> 5;
    int k4   = t % (K >> 2);
    int ct   = t / (K >> 2);
    int n = (ct << 4) + (lane & 15);
    int k = (k4 << 2) + ((lane >> 4) << 1);
    P[(size_t)idx * 2 + 0] = W[(size_t)k * N + n];
    P[(size_t)idx * 2 + 1] = W[(size_t)(k + 1) * N + n];
}

// ---------------------------------------------------------------------------
// fp32 WMMA GEMM:  Y[M,N] = A[M,K] @ W[K,N] + bias (opt. ReLU).
// K, N, RELU compile-time -> immediate-offset loads/stores throughout.
// 256-thread block = 8 waves; block owns a 16-row stripe of A staged into
// padded LDS via async-to-LDS; each wave sweeps 16x16 output tiles over N.
// Fragment layouts per CDNA5 ISA 7.12.2 (32-bit A 16x4 / C-D 16x16).
// ---------------------------------------------------------------------------
template<int K, int N, bool RELU>
__global__ void wmma_gemm_f32(const float* __restrict__ A,
                              const float* __restrict__ Bpk,
                              const float* __restrict__ bias,
                              float* __restrict__ Y) {
    extern __shared__ float As[];
    constexpr int LDSW = K + 4;        // pad -> conflict-free ds reads, 16B-mult rows
    const int tid  = threadIdx.x;
    const int lane = tid & 31;
    const int wave = tid >> 5;
    const int r0   = blockIdx.x << 4;

    // async-stage the 16 x K A-stripe (16B chunks, coalesced, ASYNCcnt-tracked)
    {
        const unsigned lb = (unsigned)(size_t)As;
        constexpr int CPR = K / 4;     // 16B chunks per row
        for (int ch = tid; ch < 16 * CPR; ch += 256) {
            const int row = ch / CPR;
            const int c4  = ch - row * CPR;
            async_ld16(lb + (unsigned)(row * LDSW + (c4 << 2)) * 4u,
                       A + (size_t)(r0 + row) * K + (c4 << 2));
        }
        wait_async0();
    }
    __syncthreads();

    const int mrow  = lane & 15;       // M row within tile (and N col within tile)
    const int hi    = lane >> 4;       // lane half
    const int khalf = hi << 1;
    const float* arow = &As[mrow * LDSW + khalf];
    const v2f* Bp = (const v2f*)Bpk;

    constexpr int NTILES = N >> 4;
#pragma unroll 1
    for (int ct = wave; ct < NTILES; ct += 8) {
        const v2f* bp = Bp + (size_t)ct * (K >> 2) * 32 + lane;
        v8f acc = {0.f, 0.f, 0.f, 0.f, 0.f, 0.f, 0.f, 0.f};
#pragma unroll 8
        for (int k4 = 0; k4 < (K >> 2); ++k4) {
            v2f a = *(const v2f*)(arow + (k4 << 2));   // ds_load_b64, imm offset
            v2f b = bp[(size_t)k4 * 32];               // global_load_b64, imm offset
            acc = __builtin_amdgcn_wmma_f32_16x16x4_f32(
                      false, a, false, b, (short)0, acc, false, false);
        }
        const int n  = (ct << 4) + mrow;
        const float bv = bias[n];
        float* Yb = Y + (size_t)(r0 + (hi << 3)) * N + n;
#pragma unroll
        for (int v = 0; v < 8; ++v) {
            float o = acc[v] + bv;
            if (RELU) o = fmaxf(o, 0.0f);
            __builtin_nontemporal_store(o, Yb + (size_t)v * N);
        }
    }
}

// ---------------------------------------------------------------------------
// agg = src  (float4 copy; self-loop term, GIN_EPS == 0)
// ---------------------------------------------------------------------------
__global__ void copy4_kernel(const float* __restrict__ src,
                             float* __restrict__ dst, int n4) {
    int i = blockIdx.x * 256 + threadIdx.x;
    if (i < n4) ((float4*)dst)[i] = ((const float4*)src)[i];
}

// ---------------------------------------------------------------------------
// agg[dst[e], :] += h[src[e], :]   (32 threads per edge, float4 per thread)
// ---------------------------------------------------------------------------
__global__ void edge_scatter(const float* __restrict__ h,
                             const int* __restrict__ esrc,
                             const int* __restrict__ edst,
                             float* __restrict__ agg) {
    long long tid = (long long)blockIdx.x * 256 + threadIdx.x;
    if (tid >= (long long)EDGES * 32) return;
    int e = (int)(tid >> 5);
    int c = ((int)tid & 31) << 2;
    int s = esrc[e], d = edst[e];
    const float4 v = *(const float4*)(h + (size_t)s * CH + c);
    float* p = agg + (size_t)d * CH + c;
    unsafeAtomicAdd(p + 0, v.x);
    unsafeAtomicAdd(p + 1, v.y);
    unsafeAtomicAdd(p + 2, v.z);
    unsafeAtomicAdd(p + 3, v.w);
}

// ---------------------------------------------------------------------------
// Column stats + BN
// ---------------------------------------------------------------------------
__global__ void zero_kernel(float* __restrict__ p, int n) {
    int i = blockIdx.x * 256 + threadIdx.x;
    if (i < n) p[i] = 0.0f;
}

__global__ void colstats(const float* __restrict__ Y,
                         float* __restrict__ sums, int M) {
    int col  = threadIdx.x & (CH - 1);
    int half = threadIdx.x >> 7;
    float s = 0.f, ss = 0.f;
    for (int r = blockIdx.x * 2 + half; r < M; r += gridDim.x * 2) {
        float v = Y[(size_t)r * CH + col];
        s += v; ss += v * v;
    }
    unsafeAtomicAdd(&sums[col], s);
    unsafeAtomicAdd(&sums[CH + col], ss);
}

__global__ void bn_finalize(const float* __restrict__ sums,
                            const float* __restrict__ g,
                            const float* __restrict__ be,
                            float* __restrict__ scsh, float invM) {
    int c = threadIdx.x;
    float mean = sums[c] * invM;
    float var  = sums[CH + c] * invM - mean * mean;
    float sc   = g[c] * rsqrtf(var + BN_EPS);
    scsh[c]      = sc;
    scsh[CH + c] = be[c] - mean * sc;
}

__global__ void bn_relu(const float* __restrict__ Y,
                        const float* __restrict__ scsh,
                        float* __restrict__ H, int n4) {
    int i = blockIdx.x * 256 + threadIdx.x;
    if (i >= n4) return;
    int c = (i << 2) & (CH - 1);
    float4 v = ((const float4*)Y)[i];
    v.x = fmaxf(v.x * scsh[c + 0] + scsh[CH + c + 0], 0.f);
    v.y = fmaxf(v.y * scsh[c + 1] + scsh[CH + c + 1], 0.f);
    v.z = fmaxf(v.z * scsh[c + 2] + scsh[CH + c + 2], 0.f);
    v.w = fmaxf(v.w * scsh[c + 3] + scsh[CH + c + 3], 0.f);
    ((float4*)H)[i] = v;
}

// ---------------------------------------------------------------------------
// feats[batch[m], it*128 + c] += H[m, c]
// ---------------------------------------------------------------------------
__global__ void pool_add(const float* __restrict__ H,
                         const int* __restrict__ batch,
                         float* __restrict__ feats, int it) {
    long long tid = (long long)blockIdx.x * 256 + threadIdx.x;
    if (tid >= (long long)NODES * 32) return;
    int m = (int)(tid >> 5);
    int c = ((int)tid & 31) << 2;
    int g = batch[m];
    float4 v = *(const float4*)(H + (size_t)m * CH + c);
    float* p = feats + (size_t)g * FDIM + it * CH + c;
    unsafeAtomicAdd(p + 0, v.x);
    unsafeAtomicAdd(p + 1, v.y);
    unsafeAtomicAdd(p + 2, v.z);
    unsafeAtomicAdd(p + 3, v.w);
}

// out[g, c] = z1[g, :] @ Wh2[:, c] + bh2[c]   (tiny: 256 x 10)
__global__ void head2_kernel(const float* __restrict__ z1,
                             const float* __restrict__ Wh2,
                             const float* __restrict__ bh2,
                             float* __restrict__ out) {
    int t = blockIdx.x * 256 + threadIdx.x;
    if (t >= NG * NCLS) return;
    int g = t / NCLS, c = t - g * NCLS;
    float s = bh2[c];
    for (int k = 0; k < FDIM; ++k)
        s += z1[(size_t)g * FDIM + k] * Wh2[(size_t)k * NCLS + c];
    out[t] = s;
}

// ---------------------------------------------------------------------------
extern "C" void kernel_launch(void* const* d_in, const int* in_sizes, int n_in,
                              void* d_out, int out_size, void* d_ws, size_t ws_size,
                              hipStream_t stream) {
    const float* X    = (const float*)d_in[0];
    const float* W1   = (const float*)d_in[1];
    const float* b1   = (const float*)d_in[2];
    const float* g1   = (const float*)d_in[3];
    const float* be1  = (const float*)d_in[4];
    const float* W2   = (const float*)d_in[5];
    const float* b2   = (const float*)d_in[6];
    const float* g2   = (const float*)d_in[7];
    const float* be2  = (const float*)d_in[8];
    const float* Wh1  = (const float*)d_in[9];
    const float* bh1  = (const float*)d_in[10];
    const float* Wh2  = (const float*)d_in[11];
    const float* bh2  = (const float*)d_in[12];
    const int* esrc   = (const int*)d_in[13];
    const int* edst   = (const int*)d_in[14];
    const int* batch  = (const int*)d_in[15];
    float* out        = (float*)d_out;

    // workspace carve-out
    float* agg   = (float*)d_ws;
    float* Y     = agg   + (size_t)NODES * CH;
    float* H     = Y     + (size_t)NODES * CH;
    float* stats = H     + (size_t)NODES * CH;    // 256 floats
    float* scsh  = stats + 256;                   // 256 floats
    float* feats = scsh  + 256;                   // NG * FDIM
    float* z1    = feats + (size_t)NG * FDIM;     // NG * FDIM
    float* wpk   = z1    + (size_t)NG * FDIM;     // 10 * CH*CH packed layer weights
    float* wh1pk = wpk   + (size_t)10 * CH * CH;  // FDIM*FDIM packed head weight

    const int n4      = NODES * CH / 4;
    const int ew_grid = (int)(((long long)EDGES * 32 + 255) / 256);
    const int nd_grid = (n4 + 255) / 256;
    const int pl_grid = (int)(((long long)NODES * 32 + 255) / 256);
    const size_t smem_node = (size_t)16 * (CH + 4) * sizeof(float);
    const size_t smem_head = (size_t)16 * (FDIM + 4) * sizeof(float);
    const float invM = 1.0f / (float)NODES;

    // one-time: repack all weights into WMMA-B fragment order
    const int rp_ch = (CH / 16) * (CH / 4) * 32;          // 8192
    for (int l = 0; l < KIT; ++l) {
        repack_B<<<(rp_ch + 255) / 256, 256, 0, stream>>>(
            W1 + (size_t)l * CH * CH, wpk + (size_t)l * CH * CH, CH, CH);
        repack_B<<<(rp_ch + 255) / 256, 256, 0, stream>>>(
            W2 + (size_t)l * CH * CH, wpk + (size_t)(KIT + l) * CH * CH, CH, CH);
    }
    const int rp_h = (FDIM / 16) * (FDIM / 4) * 32;       // 204800
    repack_B<<<(rp_h + 255) / 256, 256, 0, stream>>>(Wh1, wh1pk, FDIM, FDIM);

    zero_kernel<<<(NG * FDIM + 255) / 256, 256, 0, stream>>>(feats, NG * FDIM);

    const float* cur = X;
    for (int i = 0; i < KIT; ++i) {
        // agg = (A + I) h
        copy4_kernel<<<nd_grid, 256, 0, stream>>>(cur, agg, n4);
        edge_scatter<<<ew_grid, 256, 0, stream>>>(cur, esrc, edst, agg);

        // layer 1: Y = agg @ W1[i] + b1[i]; BN; ReLU -> H
        wmma_gemm_f32<CH, CH, false><<<NODES / 16, 256, smem_node, stream>>>(
            agg, wpk + (size_t)i * CH * CH, b1 + i * CH, Y);
        zero_kernel<<<1, 256, 0, stream>>>(stats, 256);
        colstats<<<512, 256, 0, stream>>>(Y, stats, NODES);
        bn_finalize<<<1, CH, 0, stream>>>(stats, g1 + i * CH, be1 + i * CH, scsh, invM);
        bn_relu<<<nd_grid, 256, 0, stream>>>(Y, scsh, H, n4);

        // layer 2: Y = H @ W2[i] + b2[i]; BN; ReLU -> H
        wmma_gemm_f32<CH, CH, false><<<NODES / 16, 256, smem_node, stream>>>(
            H, wpk + (size_t)(KIT + i) * CH * CH, b2 + i * CH, Y);
        zero_kernel<<<1, 256, 0, stream>>>(stats, 256);
        colstats<<<512, 256, 0, stream>>>(Y, stats, NODES);
        bn_finalize<<<1, CH, 0, stream>>>(stats, g2 + i * CH, be2 + i * CH, scsh, invM);
        bn_relu<<<nd_grid, 256, 0, stream>>>(Y, scsh, H, n4);

        // per-graph add-pool into feats[:, i*128 : (i+1)*128]
        pool_add<<<pl_grid, 256, 0, stream>>>(H, batch, feats, i);
        cur = H;
    }

    // head: z1 = relu(feats @ Wh1 + bh1) ; out = z1 @ Wh2 + bh2
    wmma_gemm_f32<FDIM, FDIM, true><<<NG / 16, 256, smem_head, stream>>>(
        feats, wh1pk, bh1, z1);
    head2_kernel<<<(NG * NCLS + 255) / 256, 256, 0, stream>>>(z1, Wh2, bh2, out);
}